// FineMatching_49151605735578
// MI455X (gfx1250) — compile-verified
//
#include <hip/hip_runtime.h>
#include <math.h>
#include <float.h>

typedef __attribute__((ext_vector_type(2))) float v2f;
typedef __attribute__((ext_vector_type(8))) float v8f;

#define WWDIM 64      // W*W
#define CDIM  128
#define RNUM  100     // WP*WP
#define NPAD  128     // 8 tiles of 16 -> 4 tiles/wave, branch-free GEMM
#define CONFS 113     // conf row stride (cols 0..111 used; pad dodges bank conflicts)
#define CHUNK 32      // K chunk (channels per LDS stage)

// One K-chunk (K=32, fully unrolled, zero control flow).
// Wave w owns N-tile tn=w and all 4 M-tiles: per K-step load 1 B fragment +
// 4 A fragments, issue 4 WMMAs into independent accumulators.
__device__ __forceinline__ void gemm_chunk32(const float* __restrict__ sA,
                                             const float* __restrict__ sB,
                                             v8f acc[4],
                                             int wave, int lm, int kh)
{
    const float* brow = sB + (wave * 16 + lm) * CHUNK + 2 * kh;
    const float* arow = sA + lm * CHUNK + 2 * kh;
    #pragma unroll
    for (int k = 0; k < 32; k += 4) {
        v2f b = *(const v2f*)(brow + k);             // shared across 4 tiles
        #pragma unroll
        for (int i = 0; i < 4; ++i) {
            v2f a = *(const v2f*)(arow + i * 16 * CHUNK + k);
            acc[i] = __builtin_amdgcn_wmma_f32_16x16x4_f32(
                false, a, false, b, (short)0, acc[i], false, false);
        }
    }
}

__global__ __launch_bounds__(256)
void fine_matching_kernel(const float* __restrict__ feat0,
                          const float* __restrict__ feat1,
                          const float* __restrict__ mk0,
                          const float* __restrict__ mk1,
                          float* __restrict__ out,
                          int M)
{
    __shared__ __align__(16) float sA[WWDIM * CHUNK];   // 64 x 32 (ch 120..127 zeroed on kc==3)
    __shared__ __align__(16) float sB[NPAD * CHUNK];    // 128 x 32 (rows 100..127 zeroed)
    __shared__ float conf[WWDIM * CONFS];               // 64 x 113
    __shared__ float maxr[WWDIM], sumr[WWDIM];
    __shared__ float maxc[NPAD], sumc[NPAD];
    __shared__ float redv[256];
    __shared__ int   redi[256];

    const int t    = threadIdx.x;
    const int lane = t & 31;
    const int wave = t >> 5;
    const int m    = blockIdx.x;

    const float* f0base = feat0 + (size_t)m * WWDIM * CDIM;
    const float* f1base = feat1 + (size_t)m * RNUM  * CDIM;

    const int lm = lane & 15;   // row/col within tile
    const int kh = lane >> 4;   // K-half select

    v8f zero = {};
    v8f acc[4];
    #pragma unroll
    for (int i = 0; i < 4; ++i) acc[i] = zero;

    // ---- batched GEMM conf_f = (f0/sqrt(C)) . (f1/sqrt(C))^T over K=120 ----
    // Uniform K=32 chunks; chunk 3 zeroes A channels 120..127 (contribution 0).
    for (int kc = 0; kc < 4; ++kc) {
        // stage sA: 64 rows x 8 float4
        #pragma unroll
        for (int i = 0; i < 2; ++i) {
            int idx = t + 256 * i;                 // 0..511
            int row = idx >> 3, c4 = idx & 7;
            float4 v = *(const float4*)(f0base + row * CDIM + kc * CHUNK + c4 * 4);
            if (kc == 3 && c4 >= 6) v = make_float4(0.f, 0.f, 0.f, 0.f); // ch 120..127
            *(float4*)(sA + row * CHUNK + c4 * 4) = v;
        }
        // stage sB rows 0..99: 800 float4
        #pragma unroll
        for (int i = 0; i < 4; ++i) {
            int idx = t + 256 * i;
            if (idx < 800) {
                int row = idx >> 3, c4 = idx & 7;
                float4 v = *(const float4*)(f1base + row * CDIM + kc * CHUNK + c4 * 4);
                *(float4*)(sB + row * CHUNK + c4 * 4) = v;
            }
        }
        // zero pad rows 100..127 (28 rows x 8 float4 = 224)
        if (t < 224) {
            int row = 100 + (t >> 3), c4 = t & 7;
            *(float4*)(sB + row * CHUNK + c4 * 4) = make_float4(0.f, 0.f, 0.f, 0.f);
        }
        // prefetch next chunk (global_prefetch_b8) to hide HBM latency
        if (kc < 3) {
            __builtin_prefetch(f0base + (t >> 2) * CDIM + (kc + 1) * CHUNK + (t & 3) * 8, 0, 0);
            __builtin_prefetch(f1base + ((t % 100)) * CDIM + (kc + 1) * CHUNK + (t >> 7) * 16, 0, 0);
        }
        __syncthreads();

        gemm_chunk32(sA, sB, acc, wave, lm, kh);
        __syncthreads();
    }

    // ---- spill conf tiles to LDS (apply 1/C scaling); wave 7 = N-pad, dropped ----
    if (wave < 7) {
        int col = wave * 16 + lm;
        #pragma unroll
        for (int i = 0; i < 4; ++i) {
            #pragma unroll
            for (int r = 0; r < 8; ++r) {
                int row = i * 16 + r + 8 * kh;
                conf[row * CONFS + col] = acc[i][r] * (1.0f / 128.0f);
            }
        }
    }
    __syncthreads();

    // ---- dual softmax statistics ----
    if (t < 64) {
        const float* row = conf + t * CONFS;
        float mx = -FLT_MAX;
        for (int r = 0; r < RNUM; ++r) mx = fmaxf(mx, row[r]);
        float s = 0.f;
        for (int r = 0; r < RNUM; ++r) s += expf(row[r] - mx);
        maxr[t] = mx; sumr[t] = s;
    } else if (t < 64 + RNUM) {
        int r = t - 64;
        float mx = -FLT_MAX;
        for (int l = 0; l < WWDIM; ++l) mx = fmaxf(mx, conf[l * CONFS + r]);
        float s = 0.f;
        for (int l = 0; l < WWDIM; ++l) s += expf(conf[l * CONFS + r] - mx);
        maxc[r] = mx; sumc[r] = s;
    }
    __syncthreads();

    // ---- sm = softmax_L * softmax_R, 10x10 -> inner 8x8, write + argmax ----
    float* out_sm = out + (size_t)m * 4096;
    float bv = -FLT_MAX; int bi = 0;
    #pragma unroll
    for (int i = 0; i < 16; ++i) {
        int f = t + 256 * i;                // 0..4095
        int l = f >> 6, p = f & 63;
        int r = ((p >> 3) + 1) * 10 + ((p & 7) + 1);
        float cv = conf[l * CONFS + r];
        float s = (expf(cv - maxr[l]) / sumr[l]) * (expf(cv - maxc[r]) / sumc[r]);
        out_sm[f] = s;
        if (s > bv) { bv = s; bi = f; }     // f increasing per thread -> first occurrence
    }
    redv[t] = bv; redi[t] = bi;
    __syncthreads();
    for (int s = 128; s > 0; s >>= 1) {
        if (t < s) {
            float v2 = redv[t + s]; int i2 = redi[t + s];
            if (v2 > redv[t] || (v2 == redv[t] && i2 < redi[t])) {
                redv[t] = v2; redi[t] = i2;
            }
        }
        __syncthreads();
    }

    // ---- fine refinement (thread 0) ----
    if (t == 0) {
        int idx   = redi[0];
        int idx_l = idx >> 6;
        int idx_r = idx & 63;

        float* out0 = out + (size_t)M * 4096;
        float* out1 = out0 + (size_t)M * 2;

        float dlx = (float)(idx_l & 7) - 4.0f + 0.5f;
        float dly = (float)(idx_l >> 3) - 4.0f + 0.5f;
        out0[2 * m + 0] = mk0[2 * m + 0] + dlx * 2.0f;
        out0[2 * m + 1] = mk0[2 * m + 1] + dly * 2.0f;

        float drx = (float)(idx_r & 7) - 4.0f + 0.5f;
        float dry = (float)(idx_r >> 3) - 4.0f + 0.5f;

        // conf_ff 3x3 patch: ff0[idx_l] . ff1[r'] / sqrt(8)
        float f0v[8];
        const float* p0 = f0base + idx_l * CDIM + (CDIM - 8);
        #pragma unroll
        for (int c = 0; c < 8; ++c) f0v[c] = p0[c];

        const float rs8 = 0.35355339059327373f; // 1/sqrt(8)
        float patch[9];
        float pmax = -FLT_MAX;
        for (int di = 0; di < 3; ++di) {
            for (int dj = 0; dj < 3; ++dj) {
                int ii = ((idx_r >> 3) + di - 1 + 10) % 10;
                int jj = ((idx_r & 7) + dj - 1 + 10) % 10;
                const float* p1 = f1base + (ii * 10 + jj) * CDIM + (CDIM - 8);
                float dot = 0.f;
                #pragma unroll
                for (int c = 0; c < 8; ++c) dot += f0v[c] * p1[c];
                float v = dot * rs8;
                patch[di * 3 + dj] = v;
                pmax = fmaxf(pmax, v);
            }
        }
        float esum = 0.f, heat[9];
        for (int k = 0; k < 9; ++k) {
            heat[k] = expf((patch[k] - pmax) * 0.1f);  // TEMP = 10
            esum += heat[k];
        }
        float ex = 0.f, ey = 0.f;
        for (int di = 0; di < 3; ++di)
            for (int dj = 0; dj < 3; ++dj) {
                float h = heat[di * 3 + dj] / esum;
                ex += h * (float)(dj - 1);
                ey += h * (float)(di - 1);
            }
        out1[2 * m + 0] = mk1[2 * m + 0] + drx * 2.0f + ex * 2.0f;
        out1[2 * m + 1] = mk1[2 * m + 1] + dry * 2.0f + ey * 2.0f;
    }
}

extern "C" void kernel_launch(void* const* d_in, const int* in_sizes, int n_in,
                              void* d_out, int out_size, void* d_ws, size_t ws_size,
                              hipStream_t stream) {
    const float* feat0 = (const float*)d_in[0];
    const float* feat1 = (const float*)d_in[1];
    const float* mk0   = (const float*)d_in[2];
    const float* mk1   = (const float*)d_in[3];
    float* out = (float*)d_out;
    int M = in_sizes[2] / 2;   // mkpts0_c has M*2 elements

    fine_matching_kernel<<<dim3(M), dim3(256), 0, stream>>>(
        feat0, feat1, mk0, mk1, out, M);
}